// DCTProcessor_53867479826579
// MI455X (gfx1250) — compile-verified
//
#include <hip/hip_runtime.h>

typedef __attribute__((ext_vector_type(2))) float v2f;
typedef __attribute__((ext_vector_type(8))) float v8f;

#define IMG_H 512
#define IMG_W 512
#define NBINS 64
#define NBC   96            // 32 batches * 3 channels
#define PATCH 16            // 16x16 patch = four 8x8 DCT blocks

// ---------------------------------------------------------------------------
// Per-wave 16x16 patch 2-D DCT via f32 WMMA.
// Dd = diag(D, D) (16x16 block-diagonal of the 8x8 DCT basis).
//   T  = Dd @ X          (X streamed from global as B operand)
//   Y' = Dd @ T^T        (T^T read from an LDS-staged tile)  => Y' = (D X D^T)^T per 8x8 block
// C/D accumulator layout (32-bit, 16x16): VGPR r, lane l -> element
//   (M = r + 8*(l>>4), N = l&15).
// A operand (16x4 f32): lane l -> M = l&15, VGPR0: K = 2*(l>>4), VGPR1: K+1.
// B operand (4x16 f32): lane l -> N = l&15, VGPR0: K = 2*(l>>4), VGPR1: K+1.
// ---------------------------------------------------------------------------
__device__ __forceinline__ v8f dct16_patch(const float* __restrict__ xp,
                                           const float* __restrict__ dct,
                                           float* __restrict__ tile, int lane)
{
    const int n  = lane & 15;   // M for A operand, N for B operand
    const int hi = lane >> 4;   // K sub-offset selector

    // A operand chunks: columns of Dd (block-diagonal => zeros off-diagonal)
    v2f A[4];
#pragma unroll
    for (int c = 0; c < 4; ++c) {
        const int k0 = 4 * c + 2 * hi;
        const int k1 = k0 + 1;
        float a0 = 0.0f, a1 = 0.0f;
        if ((n >> 3) == (k0 >> 3)) a0 = dct[(n & 7) * 8 + (k0 & 7)];
        if ((n >> 3) == (k1 >> 3)) a1 = dct[(n & 7) * 8 + (k1 & 7)];
        A[c].x = a0; A[c].y = a1;
    }

    // First GEMM: T = Dd @ X, K accumulated in 4 chunks of 4
    v8f t = {};
#pragma unroll
    for (int c = 0; c < 4; ++c) {
        const int kr = 4 * c + 2 * hi;
        v2f b;
        b.x = xp[kr * IMG_W + n];
        b.y = xp[(kr + 1) * IMG_W + n];
        t = __builtin_amdgcn_wmma_f32_16x16x4_f32(false, A[c], false, b,
                                                  (short)0, t, false, false);
    }

    // Stage T to LDS (row-major, 17-float padded rows: conflict-free transpose)
#pragma unroll
    for (int r = 0; r < 8; ++r)
        tile[(r + 8 * hi) * 17 + n] = t[r];
    // same-wave RAW through LDS: make stores visible before transposed reads
    asm volatile("s_wait_dscnt 0x0" ::: "memory");

    // Second GEMM: Y^T = Dd @ T^T ; B chunk element (kk, nn) = T[nn][kk]
    v8f y = {};
#pragma unroll
    for (int c = 0; c < 4; ++c) {
        const int kr = 4 * c + 2 * hi;
        v2f b;
        b.x = tile[n * 17 + kr];
        b.y = tile[n * 17 + kr + 1];
        y = __builtin_amdgcn_wmma_f32_16x16x4_f32(false, A[c], false, b,
                                                  (short)0, y, false, false);
    }
    return y;   // Y^T; DC coeffs sit at y[0] on lanes with (lane&7)==0
}

// ---------------------------------------------------------------------------
__global__ void dct_hist_init(unsigned* __restrict__ ws, int n)
{
    int i = blockIdx.x * blockDim.x + threadIdx.x;
    if (i < n) ws[i] = 0u;
}

__global__ void dct_hist_pass1_max(const float* __restrict__ x,
                                   const float* __restrict__ dct,
                                   unsigned* __restrict__ gmaxbits)
{
    __shared__ float tiles[8][16 * 17];
    __shared__ float wmax[8];
    const int wave = threadIdx.x >> 5;
    const int lane = threadIdx.x & 31;
    const int bc   = blockIdx.y;                 // (b*3 + c)
    const int patch = blockIdx.x * 8 + wave;     // 0..1023 within image
    const int py = patch >> 5, px = patch & 31;
    const float* xp = x + (size_t)bc * (IMG_H * IMG_W)
                        + (size_t)(py * PATCH) * IMG_W + px * PATCH;

    v8f y = dct16_patch(xp, dct, tiles[wave], lane);

    float m = 0.0f;
#pragma unroll
    for (int r = 0; r < 8; ++r) {
        float v = fabsf(y[r]);
        if (r == 0 && (lane & 7) == 0) v = 0.0f;  // drop DC of each 8x8 block
        m = fmaxf(m, v);
    }
#pragma unroll
    for (int off = 16; off > 0; off >>= 1)
        m = fmaxf(m, __shfl_xor(m, off, 32));
    if (lane == 0) wmax[wave] = m;
    __syncthreads();
    if (threadIdx.x == 0) {
        float bm = wmax[0];
#pragma unroll
        for (int i = 1; i < 8; ++i) bm = fmaxf(bm, wmax[i]);
        // mags are >= 0 -> uint bit order == float order
        atomicMax(gmaxbits, __float_as_uint(bm));
    }
}

__global__ void dct_hist_pass2_hist(const float* __restrict__ x,
                                    const float* __restrict__ dct,
                                    const unsigned* __restrict__ gmaxbits,
                                    unsigned* __restrict__ counts)
{
    __shared__ float tiles[8][16 * 17];
    __shared__ unsigned hist[NBINS];
    for (int i = threadIdx.x; i < NBINS; i += blockDim.x) hist[i] = 0u;
    __syncthreads();

    const int wave = threadIdx.x >> 5;
    const int lane = threadIdx.x & 31;
    const int bc   = blockIdx.y;
    const int patch = blockIdx.x * 8 + wave;
    const int py = patch >> 5, px = patch & 31;
    const float* xp = x + (size_t)bc * (IMG_H * IMG_W)
                        + (size_t)(py * PATCH) * IMG_W + px * PATCH;

    v8f y = dct16_patch(xp, dct, tiles[wave], lane);

    const float hist_max = __uint_as_float(*gmaxbits) * 1.1f;
    const float scale = (float)NBINS / hist_max;
#pragma unroll
    for (int r = 0; r < 8; ++r) {
        if (r == 0 && (lane & 7) == 0) continue;  // skip DC
        float v = fabsf(y[r]);
        int bin = (int)floorf(v * scale);
        bin = bin < 0 ? 0 : (bin > NBINS - 1 ? NBINS - 1 : bin);
        atomicAdd(&hist[bin], 1u);
    }
    __syncthreads();
    if (threadIdx.x < NBINS)
        atomicAdd(&counts[bc * NBINS + threadIdx.x], hist[threadIdx.x]);
}

__global__ void dct_hist_normalize(const unsigned* __restrict__ counts,
                                   float* __restrict__ out)
{
    int i = blockIdx.x * blockDim.x + threadIdx.x;
    if (i < NBC * NBINS)
        out[i] = (float)counts[i] * (1.0f / (float)(IMG_H * IMG_W));
}

// ---------------------------------------------------------------------------
extern "C" void kernel_launch(void* const* d_in, const int* in_sizes, int n_in,
                              void* d_out, int out_size, void* d_ws, size_t ws_size,
                              hipStream_t stream)
{
    const float* x   = (const float*)d_in[0];   // (32, 3, 512, 512) f32
    const float* dct = (const float*)d_in[1];   // (8, 8) f32 DCT basis
    float* out = (float*)d_out;                 // (32, 192) f32

    unsigned* ws      = (unsigned*)d_ws;
    unsigned* gmax    = ws;                     // [0]: max |coef| as uint bits
    unsigned* counts  = ws + 1;                 // [1 .. 1+96*64): histograms
    const int ws_elems = 1 + NBC * NBINS;

    dct_hist_init<<<(ws_elems + 255) / 256, 256, 0, stream>>>(ws, ws_elems);

    dim3 grid(128, NBC);                        // 128 patch-groups x 96 images
    dct_hist_pass1_max <<<grid, 256, 0, stream>>>(x, dct, gmax);
    dct_hist_pass2_hist<<<grid, 256, 0, stream>>>(x, dct, gmax, counts);

    dct_hist_normalize<<<(NBC * NBINS + 255) / 256, 256, 0, stream>>>(counts, out);
}